// make_transform_matrix_18923625906235
// MI455X (gfx1250) — compile-verified
//
#include <hip/hip_runtime.h>

// make_transform_matrix: out[b,c] = top 2 rows of R(theta) @ T(tx,ty)
//   row0 = [c, -s, c*tx - s*ty]
//   row1 = [s,  c, s*tx + c*ty]
// Elementwise stream, 36 B/element HBM traffic, AI ~ 0.22 FLOP/B.
// Roofline: ~604 MB / 23.3 TB/s ~= 26 us -> purely bandwidth-bound on MI455X.
// Design: one vec4 tile per thread (no grid-stride loop -> minimal address
// VALU), B128 non-temporal loads/stores (footprint 3x the 192 MB L2, so NT
// avoids thrashing), hardware v_sin_f32/v_cos_f32, 256-thread wave32 blocks.

typedef __attribute__((ext_vector_type(4))) float v4f;

__device__ __forceinline__ void sincos_hw(float theta, float& s, float& c) {
    // V_SIN_F32 / V_COS_F32 take input in revolutions: scale by 1/(2*pi).
    const float kInv2Pi = 0.15915494309189535f;
    float r = theta * kInv2Pi;
    s = __builtin_amdgcn_sinf(r);
    c = __builtin_amdgcn_cosf(r);
}

extern "C" __global__ __launch_bounds__(256) void
make_transform_vec4_kernel(const v4f* __restrict__ tx4,
                           const v4f* __restrict__ ty4,
                           const v4f* __restrict__ th4,
                           v4f* __restrict__ out4,
                           int n4) {
    const int i = blockIdx.x * 256 + threadIdx.x;
    if (i >= n4) return;

    // Streaming reads; th:NT so L2 isn't polluted by single-use data.
    v4f a = __builtin_nontemporal_load(tx4 + i);  // tx
    v4f b = __builtin_nontemporal_load(ty4 + i);  // ty
    v4f t = __builtin_nontemporal_load(th4 + i);  // theta

    float f[24];
#pragma unroll
    for (int j = 0; j < 4; ++j) {
        float s, c;
        sincos_hw(t[j], s, c);
        f[6 * j + 0] = c;
        f[6 * j + 1] = -s;
        f[6 * j + 2] = c * a[j] - s * b[j];  // contracts to v_fma
        f[6 * j + 3] = s;
        f[6 * j + 4] = c;
        f[6 * j + 5] = s * a[j] + c * b[j];
    }

    // 4 elements * 6 floats = 96 B contiguous per lane; wave32 covers a
    // contiguous 3 KB span. 16B-aligned. NT stores: write-once output.
    v4f* ov = out4 + (size_t)i * 6;
#pragma unroll
    for (int k = 0; k < 6; ++k) {
        v4f v = { f[4 * k + 0], f[4 * k + 1], f[4 * k + 2], f[4 * k + 3] };
        __builtin_nontemporal_store(v, ov + k);
    }
}

// Scalar tail for n % 4 != 0 (never launched for 4096x4096).
extern "C" __global__ void
make_transform_tail_kernel(const float* __restrict__ tx,
                           const float* __restrict__ ty,
                           const float* __restrict__ th,
                           float* __restrict__ out,
                           int start, int n) {
    int i = start + blockIdx.x * blockDim.x + threadIdx.x;
    if (i >= n) return;
    float s, c;
    sincos_hw(th[i], s, c);
    float* o = out + (size_t)i * 6;
    o[0] = c;
    o[1] = -s;
    o[2] = c * tx[i] - s * ty[i];
    o[3] = s;
    o[4] = c;
    o[5] = s * tx[i] + c * ty[i];
}

extern "C" void kernel_launch(void* const* d_in, const int* in_sizes, int n_in,
                              void* d_out, int out_size, void* d_ws, size_t ws_size,
                              hipStream_t stream) {
    (void)n_in; (void)out_size; (void)d_ws; (void)ws_size;
    const float* tx = (const float*)d_in[0];
    const float* ty = (const float*)d_in[1];
    const float* th = (const float*)d_in[2];
    float* out = (float*)d_out;

    const int n   = in_sizes[0];   // 16,777,216 for 4096x4096
    const int n4  = n >> 2;        // vec4 groups
    const int rem = n - (n4 << 2);

    if (n4 > 0) {
        const int threads = 256;   // 8 x wave32 per block
        const int blocks  = (n4 + threads - 1) / threads;  // 16384 blocks
        make_transform_vec4_kernel<<<blocks, threads, 0, stream>>>(
            (const v4f*)tx, (const v4f*)ty, (const v4f*)th, (v4f*)out, n4);
    }
    if (rem > 0) {
        make_transform_tail_kernel<<<1, 64, 0, stream>>>(tx, ty, th, out,
                                                         n4 << 2, n);
    }
}